// DeepND_ST_74715251081222
// MI455X (gfx1250) — compile-verified
//
#include <hip/hip_runtime.h>
#include <hip/hip_bf16.h>
#include <math.h>

#define UNIT 15
#define H1   4
#define FEAT 20
#define NEXP 16
#define BN_EPS 1e-5f

typedef float v2f __attribute__((ext_vector_type(2)));
typedef float v8f __attribute__((ext_vector_type(8)));

// ---------------------------------------------------------------------------
// init: fill a float region with a constant (deg starts at 2.0 = self loop w)
// ---------------------------------------------------------------------------
__global__ void k_fill(float* __restrict__ p, float v, long n) {
    long i = (long)blockIdx.x * blockDim.x + threadIdx.x;
    if (i < n) p[i] = v;
}

// deg -> 1/sqrt(deg)   (deg >= 2 always because of the improved self loop)
__global__ void k_dinv(float* __restrict__ p, long n) {
    long i = (long)blockIdx.x * blockDim.x + threadIdx.x;
    if (i < n) p[i] = rsqrtf(p[i]);
}

// ---------------------------------------------------------------------------
// degree: deg[e][dst] += 1 per edge  (grid.y = expert)
// ---------------------------------------------------------------------------
__global__ void k_degree(const int* __restrict__ edge, float* __restrict__ deg,
                         int N, int E) {
    int e = blockIdx.y;
    long j = (long)blockIdx.x * blockDim.x + threadIdx.x;
    if (j >= E) return;
    int d = __builtin_nontemporal_load(edge + (long)e * 2 * E + E + j);
    atomicAdd(&deg[(long)e * N + d], 1.0f);
}

// ---------------------------------------------------------------------------
// hlin1 = X @ W1 per expert via V_WMMA_F32_16X16X4_F32.
// B packs 4 experts x 4 channels into the 16 columns; grid = (tiles, 4 groups)
// One wave32 per block. All loads are unconditional with clamped addresses;
// invalid elements are zeroed via selects so no EXEC-branch diamonds appear
// and EXEC is all-ones at every WMMA.
// A 16x4 f32: lane m=lane%16, vgpr{0,1} = K{0,1} (lanes<16) / K{2,3} (lanes>=16)
// D 16x16 f32: vgpr r -> M = r + 8*(lane>=16), N = lane%16
// ---------------------------------------------------------------------------
__global__ void k_lin1_wmma(const float* __restrict__ flatten,
                            const float* __restrict__ W1,
                            float* __restrict__ hlin1, int N) {
    const int lane = threadIdx.x;
    const int tile = blockIdx.x;
    const int g    = blockIdx.y;          // expert group (4 experts)
    const int col  = lane & 15;
    const int hi   = lane >> 4;           // lane half selects K pair
    const int e    = g * 4 + (col >> 2);
    const int c    = col & 3;
    const long node = (long)tile * 16 + col;   // A-row for this lane
    const bool arow_ok = (node < N);
    const long nclamp  = arow_ok ? node : (long)(N - 1);
    const long amax    = (long)N * UNIT - 1;
    const int  bmax    = NEXP * UNIT * H1 - 1;

    v8f acc = {};
#pragma unroll
    for (int k0 = 0; k0 < UNIT; k0 += 4) {
        const int ka = k0 + 2 * hi;
        const bool k0ok = (ka < UNIT);
        const bool k1ok = (ka + 1 < UNIT);
        // A: unconditional loads from clamped addresses, then select
        long ia0 = nclamp * UNIT + ka;     if (ia0 > amax) ia0 = amax;
        long ia1 = nclamp * UNIT + ka + 1; if (ia1 > amax) ia1 = amax;
        float ta0 = flatten[ia0];
        float ta1 = flatten[ia1];
        // B: weights, uniform per lane
        int ib0 = (e * UNIT + ka) * H1 + c;     if (ib0 > bmax) ib0 = bmax;
        int ib1 = (e * UNIT + ka + 1) * H1 + c; if (ib1 > bmax) ib1 = bmax;
        float tb0 = W1[ib0];
        float tb1 = W1[ib1];
        v2f a, b;
        a.x = (arow_ok && k0ok) ? ta0 : 0.0f;
        a.y = (arow_ok && k1ok) ? ta1 : 0.0f;
        b.x = k0ok ? tb0 : 0.0f;
        b.y = k1ok ? tb1 : 0.0f;
        acc = __builtin_amdgcn_wmma_f32_16x16x4_f32(
            false, a, false, b, (short)0, acc, false, false);
    }
#pragma unroll
    for (int r = 0; r < 8; ++r) {
        long m = (long)tile * 16 + r + 8 * hi;
        if (m < N) hlin1[((long)e * N + m) * H1 + c] = acc[r];
    }
}

// ---------------------------------------------------------------------------
// layer-1 edge aggregation: agg1[e][dst] += hlin1[e][src] * dinv[s]*dinv[d]
// ---------------------------------------------------------------------------
__global__ void k_scatter1(const int* __restrict__ edge,
                           const float* __restrict__ dinv,
                           const float* __restrict__ hlin,
                           float* __restrict__ agg, int N, int E) {
    int e = blockIdx.y;
    long j = (long)blockIdx.x * blockDim.x + threadIdx.x;
    if (j >= E) return;
    const int* base = edge + (long)e * 2 * E;
    int s = __builtin_nontemporal_load(base + j);
    int d = __builtin_nontemporal_load(base + E + j);
    const float* dv = dinv + (long)e * N;
    float nrm = dv[s] * dv[d];
    float4 h = *(const float4*)(hlin + ((long)e * N + s) * 4);
    float* o = agg + ((long)e * N + d) * 4;
    atomicAdd(o + 0, h.x * nrm);
    atomicAdd(o + 1, h.y * nrm);
    atomicAdd(o + 2, h.z * nrm);
    atomicAdd(o + 3, h.w * nrm);
}

// ---------------------------------------------------------------------------
// layer-1 finalize: + self-loop + bias, relu (in place into agg1), and
// block-reduced BN statistics (sum, sumsq) -> 8 atomics per block.
// ---------------------------------------------------------------------------
__global__ void k_fin1(float* __restrict__ agg1,
                       const float* __restrict__ hlin1,
                       const float* __restrict__ dinv,
                       const float* __restrict__ b1,
                       float* __restrict__ bnstats, int N) {
    int e = blockIdx.y;
    int n = blockIdx.x * 256 + threadIdx.x;
    float h[4] = {0.f, 0.f, 0.f, 0.f};
    if (n < N) {
        float dv = dinv[(long)e * N + n];
        float sw = 2.0f * dv * dv;                    // self-loop norm (w=2)
        float* a = agg1 + ((long)e * N + n) * 4;
        const float* hl = hlin1 + ((long)e * N + n) * 4;
#pragma unroll
        for (int c = 0; c < 4; ++c) {
            h[c] = fmaxf(a[c] + hl[c] * sw + b1[e * 4 + c], 0.0f);
            a[c] = h[c];
        }
    }
    __shared__ float red[256];
#pragma unroll
    for (int q = 0; q < 8; ++q) {
        float v = (q < 4) ? h[q] : h[q - 4] * h[q - 4];
        red[threadIdx.x] = v;
        __syncthreads();
        for (int s = 128; s > 0; s >>= 1) {
            if (threadIdx.x < s) red[threadIdx.x] += red[threadIdx.x + s];
            __syncthreads();
        }
        if (threadIdx.x == 0) atomicAdd(&bnstats[e * 8 + q], red[0]);
        __syncthreads();
    }
}

// ---------------------------------------------------------------------------
// BatchNorm (batch stats, biased var) + tiny 4x2 GEMM -> hlin2
// ---------------------------------------------------------------------------
__global__ void k_bnlin2(const float* __restrict__ hrelu,   // agg1 after k_fin1
                         const float* __restrict__ bnstats,
                         const float* __restrict__ gamma,
                         const float* __restrict__ beta,
                         const float* __restrict__ W2,
                         float* __restrict__ hlin2, int N) {
    int e = blockIdx.y;
    int n = blockIdx.x * 256 + threadIdx.x;
    if (n >= N) return;
    float invN = 1.0f / (float)N;
    const float* h = hrelu + ((long)e * N + n) * 4;
    float o0 = 0.f, o1 = 0.f;
#pragma unroll
    for (int c = 0; c < 4; ++c) {
        float mu  = bnstats[e * 8 + c] * invN;
        float var = bnstats[e * 8 + 4 + c] * invN - mu * mu;
        float isd = rsqrtf(var + BN_EPS);
        float hbn = gamma[e * 4 + c] * (h[c] - mu) * isd + beta[e * 4 + c];
        o0 += hbn * W2[((long)e * 4 + c) * 2 + 0];
        o1 += hbn * W2[((long)e * 4 + c) * 2 + 1];
    }
    hlin2[((long)e * N + n) * 2 + 0] = o0;
    hlin2[((long)e * N + n) * 2 + 1] = o1;
}

// ---------------------------------------------------------------------------
// layer-2 edge aggregation (2 channels)
// ---------------------------------------------------------------------------
__global__ void k_scatter2(const int* __restrict__ edge,
                           const float* __restrict__ dinv,
                           const float* __restrict__ hlin,
                           float* __restrict__ agg, int N, int E) {
    int e = blockIdx.y;
    long j = (long)blockIdx.x * blockDim.x + threadIdx.x;
    if (j >= E) return;
    const int* base = edge + (long)e * 2 * E;
    int s = __builtin_nontemporal_load(base + j);
    int d = __builtin_nontemporal_load(base + E + j);
    const float* dv = dinv + (long)e * N;
    float nrm = dv[s] * dv[d];
    float2 h = *(const float2*)(hlin + ((long)e * N + s) * 2);
    float* o = agg + ((long)e * N + d) * 2;
    atomicAdd(o + 0, h.x * nrm);
    atomicAdd(o + 1, h.y * nrm);
}

// ---------------------------------------------------------------------------
// gate = features @ Wg^T + bg via V_WMMA_F32_16X16X4_F32 (M=16 nodes, N=16
// experts, K=20 in 5 steps). Bias preloaded into the accumulator. K=20 is a
// multiple of 4 so only the node-row tail needs masking (clamped loads).
// ---------------------------------------------------------------------------
__global__ void k_gate_wmma(const float* __restrict__ features,
                            const float* __restrict__ Wg,
                            const float* __restrict__ bg,
                            float* __restrict__ gatelin, int N) {
    const int lane = threadIdx.x;
    const int tile = blockIdx.x;
    const int col  = lane & 15;    // expert
    const int hi   = lane >> 4;
    const long node = (long)tile * 16 + col;
    const bool arow_ok = (node < N);
    const long nclamp  = arow_ok ? node : (long)(N - 1);

    float bias = bg[col];          // D(M,N=col) depends on N only
    v8f acc = {bias, bias, bias, bias, bias, bias, bias, bias};
#pragma unroll
    for (int k0 = 0; k0 < FEAT; k0 += 4) {
        const int ka = k0 + 2 * hi;
        float ta0 = features[nclamp * FEAT + ka];
        float ta1 = features[nclamp * FEAT + ka + 1];
        v2f a, b;
        a.x = arow_ok ? ta0 : 0.0f;
        a.y = arow_ok ? ta1 : 0.0f;
        b.x = Wg[(long)col * FEAT + ka];
        b.y = Wg[(long)col * FEAT + ka + 1];
        acc = __builtin_amdgcn_wmma_f32_16x16x4_f32(
            false, a, false, b, (short)0, acc, false, false);
    }
#pragma unroll
    for (int r = 0; r < 8; ++r) {
        long m = (long)tile * 16 + r + 8 * hi;
        if (m < N) gatelin[m * NEXP + col] = acc[r];
    }
}

// ---------------------------------------------------------------------------
// combine: per node, softmax gate over 16 experts, per-expert self-loop +
// bias + log_softmax over 2 classes, weighted sum -> out[N,2]
// ---------------------------------------------------------------------------
__global__ void k_combine(const float* __restrict__ agg2,
                          const float* __restrict__ hlin2,
                          const float* __restrict__ dinv,
                          const float* __restrict__ b2,
                          const float* __restrict__ gatelin,
                          float* __restrict__ out, int N) {
    int n = blockIdx.x * 256 + threadIdx.x;
    if (n >= N) return;
    float g[NEXP];
    float mx = -1e30f;
#pragma unroll
    for (int e = 0; e < NEXP; ++e) {
        g[e] = gatelin[(long)n * NEXP + e];
        mx = fmaxf(mx, g[e]);
    }
    float sum = 0.f;
#pragma unroll
    for (int e = 0; e < NEXP; ++e) { g[e] = expf(g[e] - mx); sum += g[e]; }
    float inv = 1.0f / sum;
    float acc0 = 0.f, acc1 = 0.f;
#pragma unroll
    for (int e = 0; e < NEXP; ++e) {
        float dv = dinv[(long)e * N + n];
        float sw = 2.0f * dv * dv;
        long base = ((long)e * N + n) * 2;
        float l0 = agg2[base + 0] + hlin2[base + 0] * sw + b2[e * 2 + 0];
        float l1 = agg2[base + 1] + hlin2[base + 1] * sw + b2[e * 2 + 1];
        float m = fmaxf(l0, l1);
        float z = logf(expf(l0 - m) + expf(l1 - m));
        float w = g[e] * inv;
        acc0 += w * (l0 - m - z);
        acc1 += w * (l1 - m - z);
    }
    out[(long)n * 2 + 0] = acc0;
    out[(long)n * 2 + 1] = acc1;
}

// ---------------------------------------------------------------------------
extern "C" void kernel_launch(void* const* d_in, const int* in_sizes, int n_in,
                              void* d_out, int out_size, void* d_ws, size_t ws_size,
                              hipStream_t stream) {
    const float* flatten  = (const float*)d_in[0];
    const float* features = (const float*)d_in[1];
    const int*   edge     = (const int*)d_in[2];
    const float* W1    = (const float*)d_in[3];
    const float* b1    = (const float*)d_in[4];
    const float* gamma = (const float*)d_in[5];
    const float* beta  = (const float*)d_in[6];
    const float* W2    = (const float*)d_in[7];
    const float* b2    = (const float*)d_in[8];
    const float* Wg    = (const float*)d_in[9];
    const float* bg    = (const float*)d_in[10];
    float* out = (float*)d_out;

    const int N = in_sizes[0] / UNIT;          // 25825
    const int E = in_sizes[2] / (NEXP * 2);    // 1,000,000

    float* ws = (float*)d_ws;
    size_t oDinv  = 0;
    size_t oHlin1 = oDinv  + (size_t)NEXP * N;        // 16N
    size_t oAgg1  = oHlin1 + (size_t)NEXP * N * 4;    // +64N
    size_t oHlin2 = oAgg1  + (size_t)NEXP * N * 4;    // +64N
    size_t oAgg2  = oHlin2 + (size_t)NEXP * N * 2;    // +32N
    size_t oStats = oAgg2  + (size_t)NEXP * N * 2;    // +32N
    size_t oGate  = oStats + (size_t)NEXP * 8;        // +128
    // total = 224N + 128 floats ~ 23 MB

    // init accumulators (graph-capture-safe)
    hipMemsetAsync(ws + oAgg1,  0, (size_t)NEXP * N * 4 * sizeof(float), stream);
    hipMemsetAsync(ws + oAgg2,  0, (size_t)NEXP * N * 2 * sizeof(float), stream);
    hipMemsetAsync(ws + oStats, 0, (size_t)NEXP * 8 * sizeof(float), stream);
    long nd = (long)NEXP * N;
    k_fill<<<dim3((nd + 255) / 256), 256, 0, stream>>>(ws + oDinv, 2.0f, nd);

    dim3 egrid((E + 255) / 256, NEXP);
    k_degree<<<egrid, 256, 0, stream>>>(edge, ws + oDinv, N, E);
    k_dinv<<<dim3((nd + 255) / 256), 256, 0, stream>>>(ws + oDinv, nd);

    int tiles = (N + 15) / 16;
    k_lin1_wmma<<<dim3(tiles, 4), 32, 0, stream>>>(flatten, W1, ws + oHlin1, N);
    k_scatter1<<<egrid, 256, 0, stream>>>(edge, ws + oDinv, ws + oHlin1,
                                          ws + oAgg1, N, E);
    dim3 ngrid((N + 255) / 256, NEXP);
    k_fin1<<<ngrid, 256, 0, stream>>>(ws + oAgg1, ws + oHlin1, ws + oDinv,
                                      b1, ws + oStats, N);
    k_bnlin2<<<ngrid, 256, 0, stream>>>(ws + oAgg1, ws + oStats, gamma, beta,
                                        W2, ws + oHlin2, N);
    k_scatter2<<<egrid, 256, 0, stream>>>(edge, ws + oDinv, ws + oHlin2,
                                          ws + oAgg2, N, E);
    k_gate_wmma<<<dim3(tiles), 32, 0, stream>>>(features, Wg, bg, ws + oGate, N);
    k_combine<<<dim3((N + 255) / 256), 256, 0, stream>>>(
        ws + oAgg2, ws + oHlin2, ws + oDinv, b2, ws + oGate, out, N);
}